// NuclearRepulsion_37417755082835
// MI455X (gfx1250) — compile-verified
//
#include <hip/hip_runtime.h>
#include <math.h>

// ---------------------------------------------------------------------------
// ZBL nuclear repulsion, MI455X (gfx1250, wave32).
//
// Roofline: stream nbrs (64MB) + offsets (96MB) once -> ~6.9us @ 23.3TB/s.
// All gathers (packed atom table 3.2MB, pz table 384B) are L2-resident.
// Per contributing pair: 1 v_rsq, 1 v_rcp, 5 v_exp, ~35 FMAs -> compute well
// under the memory floor. Reduction: LDS ds_add_f32 per pair, one
// global_atomic_add_f32 flush of M slots per block (<=1024 blocks).
//
// Workspace float layout:
//   [0]        inv_d
//   [1..4]     c_k / sum(c)
//   [5..8]     exponents_k * log2(e)
//   [16..111]  pz[t] = t^z_exp  (t = 0..95; z in [1,94))
//   [112..]    cumsum of num_atoms (M+1 ints)
//   [1120..]   packed atom table float4[N] {x,y,z, bits(mol<<16 | z)}
// ---------------------------------------------------------------------------

#define ZBL_CUM_OFF   112
#define ZBL_TAB_OFF   1120          // floats; byte offset 4480 (16B aligned)
#define ZBL_LOG2E     1.4426950408889634f
#define ZBL_KE        332.0636f

__device__ __forceinline__ void global_fadd_dev(float* p, float v) {
    // guaranteed hardware FP32 atomic, device scope (no CAS loop)
    asm volatile("global_atomic_add_f32 %0, %1, off scope:SCOPE_DEV"
                 :: "v"(p), "v"(v) : "memory");
}

// ---- main pair kernel (packed-table path) ---------------------------------
// LDS: pz[96] | acc[M]
__global__ __launch_bounds__(256)
void zbl_pairs_tab(const float4* __restrict__ atab,
                   const unsigned long long* __restrict__ nbrs,
                   const float* __restrict__ off,
                   const float* __restrict__ wsf,
                   float* __restrict__ out, int E, int M)
{
    extern __shared__ float smem[];
    float* s_pz  = smem;        // 96
    float* s_acc = smem + 96;   // M

    const float inv_d = wsf[0];
    const float ck0 = wsf[1], ck1 = wsf[2], ck2 = wsf[3], ck3 = wsf[4];
    const float e0  = wsf[5], e1  = wsf[6], e2  = wsf[7], e3  = wsf[8];

    for (int t = threadIdx.x; t < 96; t += blockDim.x) s_pz[t] = wsf[16 + t];
    for (int t = threadIdx.x; t < M;  t += blockDim.x) s_acc[t] = 0.0f;
    __syncthreads();

    const int stride = gridDim.x * blockDim.x;
    for (int e = blockIdx.x * blockDim.x + threadIdx.x; e < E; e += stride) {
        {   // run-ahead prefetch of the streamed arrays (global_prefetch_b8)
            int pe = e + stride * 8;
            if (pe < E) {
                __builtin_prefetch(&nbrs[pe], 0, 1);
                __builtin_prefetch(&off[3 * pe], 0, 1);
            }
        }
        // streamed, read-once -> non-temporal
        const unsigned long long nb = __builtin_nontemporal_load(&nbrs[e]);
        const int i = (int)(unsigned int)nb;
        const int j = (int)(nb >> 32);
        if (j > i) {
            const float4 ti = atab[i];              // global_load_b128 (L2 hit)
            const float4 tj = atab[j];
            const float ox = __builtin_nontemporal_load(&off[3 * e + 0]);
            const float oy = __builtin_nontemporal_load(&off[3 * e + 1]);
            const float oz = __builtin_nontemporal_load(&off[3 * e + 2]);
            const float dx = ti.x - tj.x - ox;
            const float dy = ti.y - tj.y - oy;
            const float dz = ti.z - tj.z - oz;
            const float s  = fmaf(dx, dx, fmaf(dy, dy, dz * dz)) + 3e-15f;
            // f_cut = 0 when r >= 5  <=>  s >= 25 : pair contributes nothing
            if (s < 25.0f) {
                const float rinv = __builtin_amdgcn_rsqf(s);   // v_rsq_f32
                const float r    = s * rinv;
                const int   pi   = __float_as_int(ti.w);
                const int   pj   = __float_as_int(tj.w);
                const int   zi   = pi & 0xffff;
                const int   zj   = pj & 0xffff;
                const float pzs  = s_pz[zi] + s_pz[zj];
                // u = -r/a = -r*(zi^ze+zj^ze)/d   (log2e folded into e_k)
                const float u = -(r * pzs) * inv_d;
                const float phi = fmaf(ck0, __builtin_amdgcn_exp2f(e0 * u),
                                  fmaf(ck1, __builtin_amdgcn_exp2f(e1 * u),
                                  fmaf(ck2, __builtin_amdgcn_exp2f(e2 * u),
                                       ck3 * __builtin_amdgcn_exp2f(e3 * u))));
                // f_cut = exp(-r^2/((5-r)(5+r))) = exp2(-log2e * s/(25-s))
                const float fc = __builtin_amdgcn_exp2f(
                    -ZBL_LOG2E * s * __builtin_amdgcn_rcpf(25.0f - s));
                const float pw = ZBL_KE * (float)(zi * zj) * rinv * phi * fc;
                atomicAdd(&s_acc[pi >> 16], pw);   // ds_add_f32
            }
        }
    }
    __syncthreads();
    for (int t = threadIdx.x; t < M; t += blockDim.x) {
        const float v = s_acc[t];
        if (v != 0.0f) global_fadd_dev(out + t, v);
    }
}

// ---- fallback pair kernel (no workspace table): LDS binary search ---------
// LDS: pz[96] | acc[M] | cum[M+1]
__global__ __launch_bounds__(256)
void zbl_pairs_bs(const float* __restrict__ xyz,
                  const int*   __restrict__ zt,
                  const unsigned long long* __restrict__ nbrs,
                  const float* __restrict__ off,
                  const float* __restrict__ wsf,
                  float* __restrict__ out, int E, int M)
{
    extern __shared__ float smem[];
    float* s_pz  = smem;
    float* s_acc = smem + 96;
    int*   s_cum = (int*)(smem + 96 + M);

    const float inv_d = wsf[0];
    const float ck0 = wsf[1], ck1 = wsf[2], ck2 = wsf[3], ck3 = wsf[4];
    const float e0  = wsf[5], e1  = wsf[6], e2  = wsf[7], e3  = wsf[8];

    for (int t = threadIdx.x; t < 96; t += blockDim.x) s_pz[t] = wsf[16 + t];
    for (int t = threadIdx.x; t < M;  t += blockDim.x) s_acc[t] = 0.0f;
    {
        const int* cum = ((const int*)wsf) + ZBL_CUM_OFF;
        for (int t = threadIdx.x; t <= M; t += blockDim.x) s_cum[t] = cum[t];
    }
    __syncthreads();

    const int stride = gridDim.x * blockDim.x;
    for (int e = blockIdx.x * blockDim.x + threadIdx.x; e < E; e += stride) {
        const unsigned long long nb = __builtin_nontemporal_load(&nbrs[e]);
        const int i = (int)(unsigned int)nb;
        const int j = (int)(nb >> 32);
        if (j > i) {
            const float dx = xyz[3 * i + 0] - xyz[3 * j + 0]
                           - __builtin_nontemporal_load(&off[3 * e + 0]);
            const float dy = xyz[3 * i + 1] - xyz[3 * j + 1]
                           - __builtin_nontemporal_load(&off[3 * e + 1]);
            const float dz = xyz[3 * i + 2] - xyz[3 * j + 2]
                           - __builtin_nontemporal_load(&off[3 * e + 2]);
            const float s  = fmaf(dx, dx, fmaf(dy, dy, dz * dz)) + 3e-15f;
            if (s < 25.0f) {
                const float rinv = __builtin_amdgcn_rsqf(s);
                const float r    = s * rinv;
                const int   zi   = zt[i], zj = zt[j];
                const float pzs  = s_pz[zi] + s_pz[zj];
                const float u = -(r * pzs) * inv_d;
                const float phi = fmaf(ck0, __builtin_amdgcn_exp2f(e0 * u),
                                  fmaf(ck1, __builtin_amdgcn_exp2f(e1 * u),
                                  fmaf(ck2, __builtin_amdgcn_exp2f(e2 * u),
                                       ck3 * __builtin_amdgcn_exp2f(e3 * u))));
                const float fc = __builtin_amdgcn_exp2f(
                    -ZBL_LOG2E * s * __builtin_amdgcn_rcpf(25.0f - s));
                const float pw = ZBL_KE * (float)(zi * zj) * rinv * phi * fc;
                int lo = 0, hi = M;
                while (hi - lo > 1) {
                    int mid = (lo + hi) >> 1;
                    if (s_cum[mid] <= i) lo = mid; else hi = mid;
                }
                atomicAdd(&s_acc[lo], pw);
            }
        }
    }
    __syncthreads();
    for (int t = threadIdx.x; t < M; t += blockDim.x) {
        const float v = s_acc[t];
        if (v != 0.0f) global_fadd_dev(out + t, v);
    }
}

// ---- prep: softplus params, pz table, cumsum(num_atoms), zero d_out -------
__device__ __forceinline__ float sp_f(float x) { return log1pf(expf(x)); }

__global__ void zbl_prep(const float* __restrict__ d_inv,
                         const float* __restrict__ ze_inv,
                         const float* __restrict__ c_inv,
                         const float* __restrict__ ex_inv,
                         const int*   __restrict__ num_atoms,
                         int M, float* __restrict__ wsf,
                         float* __restrict__ out, int out_n)
{
    __shared__ float s_ze;
    const int t = threadIdx.x;
    if (t == 0) {
        float d  = sp_f(d_inv[0]);
        float ze = sp_f(ze_inv[0]);
        float c[4], ex[4], cs = 0.0f;
        #pragma unroll
        for (int k = 0; k < 4; ++k) {
            c[k]  = sp_f(c_inv[k]);
            ex[k] = sp_f(ex_inv[k]);
            cs   += c[k];
        }
        wsf[0] = 1.0f / d;
        #pragma unroll
        for (int k = 0; k < 4; ++k) {
            wsf[1 + k] = c[k] / cs;
            wsf[5 + k] = ex[k] * ZBL_LOG2E;
        }
        s_ze = ze;
        int* cum = ((int*)wsf) + ZBL_CUM_OFF;
        int acc = 0;
        cum[0] = 0;
        for (int m = 0; m < M; ++m) { acc += num_atoms[m]; cum[m + 1] = acc; }
    }
    __syncthreads();
    const float ze = s_ze;
    if (t < 96)
        wsf[16 + t] = (t == 0) ? 0.0f : exp2f(ze * log2f((float)t));
    for (int o = t; o < out_n; o += blockDim.x)
        out[o] = 0.0f;
}

// ---- build packed atom table: float4 {x,y,z, bits(mol<<16 | z)} -----------
__global__ void zbl_fill_tab(const float* __restrict__ xyz,
                             const int*   __restrict__ zt,
                             const float* __restrict__ wsf,
                             float4* __restrict__ atab, int N, int M)
{
    const int* cum = ((const int*)wsf) + ZBL_CUM_OFF;
    int a = blockIdx.x * blockDim.x + threadIdx.x;
    if (a >= N) return;
    int lo = 0, hi = M;               // invariant: cum[lo] <= a < cum[hi]
    while (hi - lo > 1) {
        int mid = (lo + hi) >> 1;
        if (cum[mid] <= a) lo = mid; else hi = mid;
    }
    const int packed = (lo << 16) | (zt[a] & 0xffff);
    atab[a] = make_float4(xyz[3 * a + 0], xyz[3 * a + 1], xyz[3 * a + 2],
                          __int_as_float(packed));
}

// ---------------------------------------------------------------------------
extern "C" void kernel_launch(void* const* d_in, const int* in_sizes, int n_in,
                              void* d_out, int out_size, void* d_ws, size_t ws_size,
                              hipStream_t stream)
{
    const float* xyz    = (const float*)d_in[0];
    const int*   z      = (const int*)  d_in[1];
    const int*   nbrs   = (const int*)  d_in[2];
    const int*   natoms = (const int*)  d_in[3];
    const float* off    = (const float*)d_in[4];
    const float* d_inv  = (const float*)d_in[5];
    const float* ze_inv = (const float*)d_in[6];
    const float* c_inv  = (const float*)d_in[7];
    const float* ex_inv = (const float*)d_in[8];

    const int N = in_sizes[0] / 3;
    const int E = in_sizes[2] / 2;
    const int M = in_sizes[3];
    float* out = (float*)d_out;
    float* wsf = (float*)d_ws;

    zbl_prep<<<1, 256, 0, stream>>>(d_inv, ze_inv, c_inv, ex_inv, natoms, M,
                                    wsf, out, out_size);

    const size_t need_table =
        (size_t)ZBL_TAB_OFF * sizeof(float) + (size_t)N * sizeof(float4);
    const bool use_table = (ws_size >= need_table) && (M < 65536);
    float4* atab = (float4*)((float*)d_ws + ZBL_TAB_OFF);

    if (use_table && N > 0)
        zbl_fill_tab<<<(N + 255) / 256, 256, 0, stream>>>(xyz, z, wsf, atab, N, M);

    if (E > 0) {
        const int threads = 256;
        int blocks = (E + threads - 1) / threads;
        if (blocks > 1024) blocks = 1024;   // cap: flush atomics <= 1024*M
        if (use_table) {
            const size_t smem = (size_t)(96 + M) * sizeof(float);
            zbl_pairs_tab<<<blocks, threads, smem, stream>>>(
                atab, (const unsigned long long*)nbrs, off, wsf, out, E, M);
        } else {
            const size_t smem = (size_t)(96 + 2 * M + 1) * sizeof(float);
            zbl_pairs_bs<<<blocks, threads, smem, stream>>>(
                xyz, z, (const unsigned long long*)nbrs, off, wsf, out, E, M);
        }
    }
}